// FNO_RNN_2d_36919538877136
// MI455X (gfx1250) — compile-verified
//
#include <hip/hip_runtime.h>
#include <hip/hip_bf16.h>
#include <math.h>
#include <cstdint>

// FNO-RNN 2D for MI455X (gfx1250): all dense work on V_WMMA_F32_16X16X4_F32,
// LDS staging via GLOBAL_LOAD_ASYNC_TO_LDS (ASYNCcnt) where the copy is a
// straight memcpy. B=8, X=128, Y=128, CIN=2, COUT=1, W=24, D=4, T=8, YH=65.

typedef float v2f __attribute__((ext_vector_type(2)));
typedef float v8f __attribute__((ext_vector_type(8)));

__device__ __forceinline__ v8f zero8() {
  v8f z;
#pragma unroll
  for (int i = 0; i < 8; ++i) z[i] = 0.f;
  return z;
}

// D(16x16) = A(16x4) * B(4x16) + C, fp32 WMMA.
__device__ __forceinline__ v8f wmma4(v2f a, v2f b, v8f c) {
  return __builtin_amdgcn_wmma_f32_16x16x4_f32(false, a, false, b, (short)0, c,
                                               false, false);
}

// Async global->LDS copies (CDNA5, tracked with ASYNCcnt). The LDS address is
// the low 32 bits of the flat shared pointer.
__device__ __forceinline__ void async_b128(void* lds, const void* g) {
  uint32_t l = (uint32_t)(uintptr_t)lds;
  asm volatile("global_load_async_to_lds_b128 %0, %1, off" ::"v"(l), "v"(g)
               : "memory");
}
__device__ __forceinline__ void async_b64(void* lds, const void* g) {
  uint32_t l = (uint32_t)(uintptr_t)lds;
  asm volatile("global_load_async_to_lds_b64 %0, %1, off" ::"v"(l), "v"(g)
               : "memory");
}
__device__ __forceinline__ void async_wait0() {
  asm volatile("s_wait_asynccnt 0" ::: "memory");
}

// ---------------- workspace layout (float offsets) ----------------
static constexpr size_t N_H   = (size_t)8 * 128 * 128 * 24;   // 3,145,728
static constexpr size_t N_SPC = (size_t)8 * 128 * 65 * 24;    // 1,597,440

static constexpr size_t OFF_FYR  = 0;                          // 80x128 (padded)
static constexpr size_t OFF_FYI  = OFF_FYR  + 80 * 128;
static constexpr size_t OFF_FXR  = OFF_FYI  + 80 * 128;        // 128x128
static constexpr size_t OFF_FXI  = OFF_FXR  + 128 * 128;
static constexpr size_t OFF_IFXR = OFF_FXI  + 128 * 128;
static constexpr size_t OFF_IFXI = OFF_IFXR + 128 * 128;
static constexpr size_t OFF_GYR  = OFF_IFXI + 128 * 128;       // 128x68 (padded)
static constexpr size_t OFF_GYI  = OFF_GYR  + 128 * 68;
static constexpr size_t OFF_HA   = OFF_GYI  + 128 * 68;
static constexpr size_t OFF_HB   = OFF_HA + N_H;
static constexpr size_t OFF_PR   = OFF_HB + N_H;
static constexpr size_t OFF_PI   = OFF_PR + N_SPC;
static constexpr size_t OFF_QR   = OFF_PI + N_SPC;
static constexpr size_t OFF_QI   = OFF_QR + N_SPC;
// total ~12.78M floats ~= 49 MB of d_ws

// ---------------- DFT table construction (zero-padded) ----------------
__global__ void k_init_dft(float* Fyr, float* Fyi, float* Fxr, float* Fxi,
                           float* iFxr, float* iFxi, float* Gyr, float* Gyi) {
  int tid = blockIdx.x * 256 + threadIdx.x;
  const float TWO_PI_N = 6.28318530717958647692f / 128.0f;
  if (tid < 128 * 128) {
    int i = tid >> 7, j = tid & 127;
    float ang = TWO_PI_N * (float)((i * j) & 127);
    float s, c;
    __sincosf(ang, &s, &c);
    Fxr[tid] = c;
    Fxi[tid] = -s;
    iFxr[tid] = c * (1.f / 128.f);
    iFxi[tid] = s * (1.f / 128.f);
  }
  if (tid < 80 * 128) {  // Fy[f][y] = exp(-2*pi*i*f*y/128); rows 65..79 zero
    int f = tid >> 7, y = tid & 127;
    float vr = 0.f, vi = 0.f;
    if (f < 65) {
      float ang = TWO_PI_N * (float)((f * y) & 127);
      float s, c;
      __sincosf(ang, &s, &c);
      vr = c;
      vi = -s;
    }
    Fyr[tid] = vr;
    Fyi[tid] = vi;
  }
  if (tid < 128 * 68) {  // Gy[y][f] stride 68: hermitian-folded inverse rDFT
    int y = tid / 68, f = tid % 68;
    float vr = 0.f, vi = 0.f;
    if (f < 65) {
      float cf = (f == 0 || f == 64) ? 1.f : 2.f;
      float ang = TWO_PI_N * (float)((f * y) & 127);
      float s, c;
      __sincosf(ang, &s, &c);
      vr = cf * c * (1.f / 128.f);
      vi = -cf * s * (1.f / 128.f);
    }
    Gyr[tid] = vr;
    Gyi[tid] = vi;
  }
}

// ---------------- input mapping: h0 = x @ in_w + in_b ----------------
__global__ void k_in_map(const float* __restrict__ x,
                         const float* __restrict__ in_w,
                         const float* __restrict__ in_b,
                         float* __restrict__ h) {
  size_t idx = (size_t)blockIdx.x * 256 + threadIdx.x;
  if (idx >= N_H) return;
  int w = (int)(idx % 24);
  size_t s = idx / 24;
  h[idx] = x[s * 2] * in_w[w] + x[s * 2 + 1] * in_w[24 + w] + in_b[w];
}

// ---------------- pass A: rDFT along Y as WMMA GEMM ----------------
// per (b,x): C(65x24) = Fy(80x128 padded) @ h_slice(128x24), re & im.
__global__ void __launch_bounds__(64) k_dft_y(const float* __restrict__ h,
                                              const float* __restrict__ Fr,
                                              const float* __restrict__ Fi,
                                              float* __restrict__ Pr,
                                              float* __restrict__ Pi) {
  __shared__ __align__(16) float hs[128 * 24 + 40];  // pad: N-overhang reads
  int bx = blockIdx.x;  // b*128+x
  const float* hsrc = h + (size_t)bx * 3072;
  for (int j = threadIdx.x; j < 192; j += 64)
    async_b128(hs + j * 4, hsrc + j * 4);
  async_wait0();
  __syncthreads();
  int wave = threadIdx.x >> 5;  // 0 -> re, 1 -> im
  int lane = threadIdx.x & 31;
  int ml = lane & 15, kh = lane >> 4;
  const float* F = wave ? Fi : Fr;
  float* obase = (wave ? Pi : Pr) + (size_t)bx * 1560;
  for (int mt = 0; mt < 5; ++mt) {
    int f = mt * 16 + ml;  // < 80, table padded
    for (int nt = 0; nt < 2; ++nt) {
      int n = nt * 16 + ml;
      bool nv = n < 24;
      v8f acc = zero8();
#pragma unroll 4
      for (int ks = 0; ks < 32; ++ks) {
        int k0 = ks * 4 + kh * 2;
        v2f a, b;
        a.x = F[f * 128 + k0];
        a.y = F[f * 128 + k0 + 1];
        b.x = hs[k0 * 24 + n];
        b.y = hs[(k0 + 1) * 24 + n];
        acc = wmma4(a, b, acc);
      }
      if (nv) {
#pragma unroll
        for (int r = 0; r < 8; ++r) {
          int fr = mt * 16 + r + 8 * kh;
          if (fr < 65) obase[fr * 24 + n] = acc[r];
        }
      }
    }
  }
}

// ---------------- pass B/D: complex GEMM along X ----------------
// per b: C(128 x 1560) = A(128x128, complex) @ B(128 x 1560, complex)
__global__ void __launch_bounds__(128) k_cgemm_x(
    const float* __restrict__ Ar, const float* __restrict__ Ai,
    const float* __restrict__ Br, const float* __restrict__ Bi,
    float* __restrict__ Cr, float* __restrict__ Ci) {
  __shared__ __align__(16) float Asr[16 * 132], Asi[16 * 132];
  int b = blockIdx.z, mt = blockIdx.y, ng = blockIdx.x;
  for (int j = threadIdx.x; j < 1024; j += 128) {
    int sel = j >> 9, m = (j >> 5) & 15, kc = j & 31;
    const float* g = (sel ? Ai : Ar) + (mt * 16 + m) * 128 + kc * 4;
    float* l = (sel ? Asi : Asr) + m * 132 + kc * 4;
    async_b128(l, g);
  }
  async_wait0();
  __syncthreads();
  int wave = threadIdx.x >> 5, lane = threadIdx.x & 31;
  int ml = lane & 15, kh = lane >> 4;
  int n = (ng * 4 + wave) * 16 + ml;
  bool nv = n < 1560;
  int nc = nv ? n : 0;  // address clamp instead of predicated loads
  const float* Bbr = Br + (size_t)b * 128 * 1560;
  const float* Bbi = Bi + (size_t)b * 128 * 1560;
  v8f cr = zero8(), ci = zero8();
#pragma unroll 2
  for (int ks = 0; ks < 32; ++ks) {
    int k0 = ks * 4 + kh * 2;
    v2f ar, ai, ain, br, bi;
    ar.x = Asr[ml * 132 + k0];
    ar.y = Asr[ml * 132 + k0 + 1];
    ai.x = Asi[ml * 132 + k0];
    ai.y = Asi[ml * 132 + k0 + 1];
    ain.x = -ai.x;  // f32 WMMA has no A-negate modifier; negate in VALU
    ain.y = -ai.y;
    br.x = Bbr[(size_t)k0 * 1560 + nc];
    br.y = Bbr[(size_t)(k0 + 1) * 1560 + nc];
    bi.x = Bbi[(size_t)k0 * 1560 + nc];
    bi.y = Bbi[(size_t)(k0 + 1) * 1560 + nc];
    cr = wmma4(ar, br, cr);
    cr = wmma4(ain, bi, cr);
    ci = wmma4(ar, bi, ci);
    ci = wmma4(ai, br, ci);
  }
  if (nv) {
    float* ocr = Cr + (size_t)b * 128 * 1560;
    float* oci = Ci + (size_t)b * 128 * 1560;
#pragma unroll
    for (int r = 0; r < 8; ++r) {
      int m = mt * 16 + r + 8 * kh;
      ocr[(size_t)m * 1560 + n] = cr[r];
      oci[(size_t)m * 1560 + n] = ci[r];
    }
  }
}

// ---------------- pass C: per-frequency spectral matmul ----------------
// 13 frequencies per workgroup (65 = 5*13, no OOB anywhere).
// per frequency (kx,f): Y(8x24) = H(8x24) @ Wc(24x24) complex, via
// A = [Hre; Him] (16x24):  P = A@Wre, Q = A@Wim
//   Yre = P[0:8] - Q[8:16],  Yim = Q[0:8] + P[8:16]  (shfl_xor 16 recombine)
#define FT 13
__global__ void __launch_bounds__(128) k_spectral(
    const float* __restrict__ Hr, const float* __restrict__ Hi,
    const float* __restrict__ wsp,  // spec_w + layer offset
    float* __restrict__ Yr, float* __restrict__ Yi) {
  // lw[pair = win*24+wout][f*2+c], stride 26; pad for N-overhang reads
  __shared__ __align__(16) float lw[576 * 26 + 256];
  int kx = blockIdx.x, fbase = blockIdx.y * FT;
  for (int j = threadIdx.x; j < 576 * 13; j += 128) {
    int pair = j / 13, q = j - pair * 13;
    const float* g =
        wsp + (size_t)pair * 16640 + (size_t)kx * 130 + (size_t)(fbase + q) * 2;
    async_b64(lw + pair * 26 + q * 2, g);
  }
  async_wait0();
  __syncthreads();
  int wave = threadIdx.x >> 5, lane = threadIdx.x & 31;
  int ml = lane & 15, kh = lane >> 4;
  for (int fl = wave; fl < FT; fl += 4) {
    int f = fbase + fl;
    // A fragments: rows 0-7 = Hre(b), rows 8-15 = Him(b)
    v2f a[6];
    const float* src = (ml < 8) ? (Hr + (size_t)ml * 128 * 1560)
                                : (Hi + (size_t)(ml - 8) * 128 * 1560);
    size_t abase = ((size_t)kx * 65 + f) * 24;
#pragma unroll
    for (int ks = 0; ks < 6; ++ks) {
      int k0 = ks * 4 + kh * 2;
      a[ks].x = src[abase + k0];
      a[ks].y = src[abase + k0 + 1];
    }
    for (int nt = 0; nt < 2; ++nt) {
      int n = nt * 16 + ml;
      bool nv = n < 24;
      v8f P = zero8(), Q = zero8();
#pragma unroll
      for (int ks = 0; ks < 6; ++ks) {
        int k0 = ks * 4 + kh * 2;
        v2f br, bi;
        br.x = lw[(k0 * 24 + n) * 26 + fl * 2];
        br.y = lw[((k0 + 1) * 24 + n) * 26 + fl * 2];
        bi.x = lw[(k0 * 24 + n) * 26 + fl * 2 + 1];
        bi.y = lw[((k0 + 1) * 24 + n) * 26 + fl * 2 + 1];
        P = wmma4(a[ks], br, P);
        Q = wmma4(a[ks], bi, Q);
      }
#pragma unroll
      for (int r = 0; r < 8; ++r) {
        float qs = __shfl_xor(Q[r], 16, 32);  // bring Him@Wim row r
        float ps = __shfl_xor(P[r], 16, 32);  // bring Him@Wre row r
        if (lane < 16 && nv) {
          size_t o = ((size_t)r * 128 * 65 + (size_t)kx * 65 + f) * 24 + n;
          Yr[o] = P[r] - qs;
          Yi[o] = Q[r] + ps;
        }
      }
    }
  }
}

// ---------------- pass E: inverse-Y DFT fused with relu + 1x1 conv ----------------
// per (b,x): h' = relu(Gyr@Gfr + Gyi@Gfi) + h@cw + cb
__global__ void __launch_bounds__(128) k_ifft_y_fuse(
    const float* __restrict__ Gr, const float* __restrict__ Gi,
    const float* __restrict__ Gyr, const float* __restrict__ Gyi,  // 128x68 pad
    const float* __restrict__ hin, const float* __restrict__ cw,
    const float* __restrict__ cb, float* __restrict__ hout) {
  __shared__ __align__(16) float gr_s[1672], gi_s[1672];  // 65x24 + zero K-pad
  __shared__ __align__(16) float h_s[128 * 24 + 40];
  __shared__ __align__(16) float cw_s[616];
  __shared__ float cb_s[24];
  int bx = blockIdx.x;
  {
    const float* grs = Gr + (size_t)bx * 1560;
    const float* gis = Gi + (size_t)bx * 1560;
    const float* hsrc = hin + (size_t)bx * 3072;
    for (int j = threadIdx.x; j < 390; j += 128) {
      async_b128(gr_s + j * 4, grs + j * 4);
      async_b128(gi_s + j * 4, gis + j * 4);
    }
    for (int j = threadIdx.x; j < 768; j += 128)
      async_b128(h_s + j * 4, hsrc + j * 4);
    for (int j = threadIdx.x; j < 144; j += 128)
      async_b128(cw_s + j * 4, cw + j * 4);
    if (threadIdx.x < 112) {  // zero the K-padding rows (feed stored columns)
      gr_s[1560 + threadIdx.x] = 0.f;
      gi_s[1560 + threadIdx.x] = 0.f;
    }
    if (threadIdx.x < 24) cb_s[threadIdx.x] = cb[threadIdx.x];
  }
  async_wait0();
  __syncthreads();
  int wave = threadIdx.x >> 5, lane = threadIdx.x & 31;
  int ml = lane & 15, kh = lane >> 4;
  float* ob = hout + (size_t)bx * 3072;
  for (int job = wave; job < 16; job += 4) {
    int mt = job >> 1, nt = job & 1;
    int m = mt * 16 + ml;
    int n = nt * 16 + ml;
    bool nv = n < 24;
    v8f acc1 = zero8(), acc2 = zero8();
#pragma unroll 2
    for (int ks = 0; ks < 17; ++ks) {  // K=65 padded to 68 (zeros)
      int k0 = ks * 4 + kh * 2;
      v2f ar, ai, b1r, b1i;
      ar.x = Gyr[m * 68 + k0];
      ar.y = Gyr[m * 68 + k0 + 1];
      ai.x = Gyi[m * 68 + k0];
      ai.y = Gyi[m * 68 + k0 + 1];
      b1r.x = gr_s[k0 * 24 + n];
      b1r.y = gr_s[(k0 + 1) * 24 + n];
      b1i.x = gi_s[k0 * 24 + n];
      b1i.y = gi_s[(k0 + 1) * 24 + n];
      acc1 = wmma4(ar, b1r, acc1);
      acc1 = wmma4(ai, b1i, acc1);
    }
#pragma unroll
    for (int ks = 0; ks < 6; ++ks) {  // 1x1 conv, K=24
      int k0 = ks * 4 + kh * 2;
      v2f a2, b2;
      a2.x = h_s[m * 24 + k0];
      a2.y = h_s[m * 24 + k0 + 1];
      b2.x = cw_s[k0 * 24 + n];
      b2.y = cw_s[(k0 + 1) * 24 + n];
      acc2 = wmma4(a2, b2, acc2);
    }
    if (nv) {
#pragma unroll
      for (int r = 0; r < 8; ++r) {
        int y = mt * 16 + r + 8 * kh;
        ob[y * 24 + n] = fmaxf(acc1[r], 0.f) + acc2[r] + cb_s[n];
      }
    }
  }
}

// ---------------- output mapping: y = h @ out_w + out_b (COUT=1) ----------------
__global__ void k_out(const float* __restrict__ h, const float* __restrict__ ow,
                      const float* __restrict__ obias, float* __restrict__ out,
                      int t) {
  int s = blockIdx.x * 256 + threadIdx.x;  // over B*X*Y = 131072
  if (s >= 8 * 128 * 128) return;
  const float* hp = h + (size_t)s * 24;
  float acc = obias[0];
#pragma unroll
  for (int w = 0; w < 24; ++w) acc += hp[w] * ow[w];
  int b = s >> 14, r = s & 16383;
  out[((size_t)b * 8 + t) * 16384 + r] = acc;
}

extern "C" void kernel_launch(void* const* d_in, const int* in_sizes, int n_in,
                              void* d_out, int out_size, void* d_ws,
                              size_t ws_size, hipStream_t stream) {
  (void)in_sizes; (void)n_in; (void)out_size; (void)ws_size;
  const float* x      = (const float*)d_in[0];
  const float* in_w   = (const float*)d_in[1];
  const float* in_b   = (const float*)d_in[2];
  const float* spec_w = (const float*)d_in[3];
  const float* conv_w = (const float*)d_in[4];
  const float* conv_b = (const float*)d_in[5];
  const float* out_w  = (const float*)d_in[6];
  const float* out_b  = (const float*)d_in[7];
  // d_in[8] = num_time_steps (device scalar; T=8 compiled in, matches reference)
  float* out = (float*)d_out;
  float* ws = (float*)d_ws;

  float* Fyr = ws + OFF_FYR;   float* Fyi = ws + OFF_FYI;
  float* Fxr = ws + OFF_FXR;   float* Fxi = ws + OFF_FXI;
  float* iFxr = ws + OFF_IFXR; float* iFxi = ws + OFF_IFXI;
  float* Gyr = ws + OFF_GYR;   float* Gyi = ws + OFF_GYI;
  float* hA = ws + OFF_HA;     float* hB = ws + OFF_HB;
  float* Pr = ws + OFF_PR;     float* Pi = ws + OFF_PI;
  float* Qr = ws + OFF_QR;     float* Qi = ws + OFF_QI;

  k_init_dft<<<64, 256, 0, stream>>>(Fyr, Fyi, Fxr, Fxi, iFxr, iFxi, Gyr, Gyi);
  k_in_map<<<(int)((N_H + 255) / 256), 256, 0, stream>>>(x, in_w, in_b, hA);

  float* cur = hA;
  float* nxt = hB;
  for (int t = 0; t < 8; ++t) {
    for (int i = 0; i < 4; ++i) {
      k_dft_y<<<1024, 64, 0, stream>>>(cur, Fyr, Fyi, Pr, Pi);
      k_cgemm_x<<<dim3(25, 8, 8), 128, 0, stream>>>(Fxr, Fxi, Pr, Pi, Qr, Qi);
      k_spectral<<<dim3(128, 5), 128, 0, stream>>>(
          Qr, Qi, spec_w + (size_t)i * 24 * 24 * 128 * 65 * 2, Pr, Pi);
      k_cgemm_x<<<dim3(25, 8, 8), 128, 0, stream>>>(iFxr, iFxi, Pr, Pi, Qr, Qi);
      k_ifft_y_fuse<<<1024, 128, 0, stream>>>(Qr, Qi, Gyr, Gyi, cur,
                                              conv_w + (size_t)i * 576,
                                              conv_b + (size_t)i * 24, nxt);
      float* tmp = cur; cur = nxt; nxt = tmp;
    }
    k_out<<<(131072 + 255) / 256, 256, 0, stream>>>(cur, out_w, out_b, out, t);
  }
}